// TopKGate_33414845562944
// MI455X (gfx1250) — compile-verified
//
#include <hip/hip_runtime.h>
#include <cstdint>

// ---------------------------------------------------------------------------
// Top-1 MoE gate (TopKGate, k=1): S=8192 tokens, M=2048 model dim, E=64
// experts, capacity = ceil(S/E) = 128.
// Outputs (flat, in order): l_aux [1], combine_weights [S*E*C],
// dispatch_mask [S*E*C] (as 0/1 float), exp_counts [E].
// Cost model (MI455X): 536 MB output zero-fill dominates (~23 us @ 23.3TB/s).
// GEMM is 2.1 GFLOP / 67 MB read -> use exact f32 WMMA (16x16x4).
// ---------------------------------------------------------------------------

typedef __attribute__((ext_vector_type(2))) float v2f;
typedef __attribute__((ext_vector_type(8))) float v8f;

#define S_TOK 8192
#define MDIM  2048
#define NEXP  64
#define CAP   128
#define GATE_BLOCKS (S_TOK / 256)

// ---------------------------------------------------------------------------
// Kernel 1: logits[S,E] = inputs[S,M] @ wg[E,M]^T  via V_WMMA_F32_16X16X4_F32.
// One wave handles a 16-token x 64-expert tile: 4 accumulators share operand A.
// A layout (16x4 f32): lane<16 holds A[M=lane][{k,k+1}], lane>=16 A[M][{k+2,k+3}].
// B layout (4x16 f32): lane<16 holds B[{k,k+1}][N=lane] = wg[N][{k,k+1}] -> same
// per-lane addressing pattern as A (both matrices are row-major over K).
// ---------------------------------------------------------------------------
__global__ __launch_bounds__(256) void k_gemm(const float* __restrict__ inp,
                                              const float* __restrict__ wg,
                                              float* __restrict__ logits) {
  const int wave = threadIdx.x >> 5;
  const int lane = threadIdx.x & 31;
  const int half = lane >> 4;   // 0 -> K pair {k,k+1}; 1 -> {k+2,k+3}
  const int l16  = lane & 15;
  const int s0   = (blockIdx.x * 8 + wave) * 16;

  const float* a_row  = inp + (size_t)(s0 + l16) * MDIM + (half << 1);
  const float* b_row0 = wg  + (size_t)(l16 +  0) * MDIM + (half << 1);
  const float* b_row1 = wg  + (size_t)(l16 + 16) * MDIM + (half << 1);
  const float* b_row2 = wg  + (size_t)(l16 + 32) * MDIM + (half << 1);
  const float* b_row3 = wg  + (size_t)(l16 + 48) * MDIM + (half << 1);

  v8f acc0 = {}, acc1 = {}, acc2 = {}, acc3 = {};
#pragma unroll 4
  for (int k = 0; k < MDIM; k += 4) {
    v2f a  = *(const v2f*)(a_row  + k);
    v2f b0 = *(const v2f*)(b_row0 + k);
    v2f b1 = *(const v2f*)(b_row1 + k);
    v2f b2 = *(const v2f*)(b_row2 + k);
    v2f b3 = *(const v2f*)(b_row3 + k);
    // (neg_a, A, neg_b, B, c_mod, C, reuse_a, reuse_b)
    acc0 = __builtin_amdgcn_wmma_f32_16x16x4_f32(false, a, false, b0, (short)0, acc0, false, false);
    acc1 = __builtin_amdgcn_wmma_f32_16x16x4_f32(false, a, false, b1, (short)0, acc1, false, false);
    acc2 = __builtin_amdgcn_wmma_f32_16x16x4_f32(false, a, false, b2, (short)0, acc2, false, false);
    acc3 = __builtin_amdgcn_wmma_f32_16x16x4_f32(false, a, false, b3, (short)0, acc3, false, false);
  }

  // C/D layout: VGPR r -> M=r (lanes 0-15), M=8+r (lanes 16-31); N = lane%16.
  const int rbase = s0 + (half << 3);
#pragma unroll
  for (int r = 0; r < 8; ++r) {
    float* o = logits + (size_t)(rbase + r) * NEXP + l16;
    o[0]  = acc0[r];
    o[16] = acc1[r];
    o[32] = acc2[r];
    o[48] = acc3[r];
  }
}

// ---------------------------------------------------------------------------
// Kernel 2: per-token softmax over E=64 -> argmax index, gate value at argmax
// (= 1/sum(exp(l-max))), plus per-block partial column sums of gates and
// per-block argmax histograms (deterministic l_aux reduction later).
// ---------------------------------------------------------------------------
__global__ __launch_bounds__(256) void k_gate(const float* __restrict__ logits,
                                              int* __restrict__ idx,
                                              float* __restrict__ gateval,
                                              float* __restrict__ pbsum,
                                              int* __restrict__ pbcnt) {
  __shared__ float lsum[NEXP];
  __shared__ int   lcnt[NEXP];
  const int t = threadIdx.x;
  if (t < NEXP) { lsum[t] = 0.0f; lcnt[t] = 0; }
  __syncthreads();

  const int s = blockIdx.x * 256 + t;
  const float* row = logits + (size_t)s * NEXP;

  float mx = row[0];
  int   am = 0;
#pragma unroll 8
  for (int e = 1; e < NEXP; ++e) {
    float v = row[e];
    if (v > mx) { mx = v; am = e; }   // first-occurrence argmax (matches jnp)
  }
  float sum = 0.0f;
#pragma unroll 8
  for (int e = 0; e < NEXP; ++e) sum += __expf(row[e] - mx);
  const float inv = 1.0f / sum;

  idx[s]     = am;
  gateval[s] = inv;                    // softmax value at the argmax expert

  atomicAdd(&lcnt[am], 1);
#pragma unroll 8
  for (int e = 0; e < NEXP; ++e)
    atomicAdd(&lsum[e], __expf(row[e] - mx) * inv);
  __syncthreads();

  if (t < NEXP) {
    pbsum[blockIdx.x * NEXP + t] = lsum[t];
    pbcnt[blockIdx.x * NEXP + t] = lcnt[t];
  }
}

// ---------------------------------------------------------------------------
// Kernel 3: per-expert sequential position (cumsum over tokens). Serial in S,
// so run one workgroup: stage 1024 indices in LDS, then 64 "expert" threads
// scan them with a private running counter. loc[s] = 0-based arrival order of
// token s at its expert.
// ---------------------------------------------------------------------------
__global__ __launch_bounds__(256) void k_loc(const int* __restrict__ idx,
                                             int* __restrict__ loc) {
  __shared__ int sidx[1024];
  const int t = threadIdx.x;
  int count = 0;
  for (int base = 0; base < S_TOK; base += 1024) {
    for (int i = t; i < 1024; i += 256) sidx[i] = idx[base + i];
    __syncthreads();
    if (t < NEXP) {
#pragma unroll 8
      for (int i = 0; i < 1024; ++i)
        if (sidx[i] == t) loc[base + i] = count++;
    }
    __syncthreads();
  }
}

// ---------------------------------------------------------------------------
// Kernel 4: zero-fill the 536 MB output region with float4 stores (the
// bandwidth-dominant step). Tail element handled by thread 0.
// ---------------------------------------------------------------------------
__global__ __launch_bounds__(256) void k_fill(float4* __restrict__ p,
                                              float* __restrict__ ps,
                                              long long n4, long long ntotal) {
  const long long i = (long long)blockIdx.x * blockDim.x + threadIdx.x;
  if (i < n4) p[i] = make_float4(0.0f, 0.0f, 0.0f, 0.0f);
  if (i == 0) {
    for (long long j = n4 * 4; j < ntotal; ++j) ps[j] = 0.0f;
  }
}

// ---------------------------------------------------------------------------
// Kernel 5: scatter the 8192 nonzeros into combine_weights / dispatch_mask,
// write exp_counts, and reduce partial sums into l_aux deterministically.
// ---------------------------------------------------------------------------
__global__ __launch_bounds__(256) void k_finalize(const int* __restrict__ idx,
                                                  const int* __restrict__ loc,
                                                  const float* __restrict__ gateval,
                                                  const float* __restrict__ pbsum,
                                                  const int* __restrict__ pbcnt,
                                                  float* __restrict__ out) {
  const size_t SEC = (size_t)S_TOK * NEXP * CAP;
  const int t = threadIdx.x;
  const int s = blockIdx.x * 256 + t;

  if (s < S_TOK) {
    const int e = idx[s];
    const int l = loc[s];
    if (l < CAP) {                       // within capacity -> kept
      const float g = gateval[s];
      const size_t o = 1 + (size_t)s * (NEXP * CAP) + (size_t)e * CAP + (size_t)l;
      out[o]       = g;                  // combine_weights[s, e, l]
      out[o + SEC] = 1.0f;               // dispatch_mask (g > 0 always)
    }
  }

  if (blockIdx.x == 0) {
    __shared__ float sg[NEXP];
    __shared__ int   sc[NEXP];
    if (t < NEXP) {
      float a = 0.0f; int c = 0;
      for (int b = 0; b < GATE_BLOCKS; ++b) {
        a += pbsum[b * NEXP + t];
        c += pbcnt[b * NEXP + t];
      }
      sg[t] = a; sc[t] = c;
      out[1 + 2 * SEC + t] = (float)c;   // exp_counts (pre-capacity)
    }
    __syncthreads();
    if (t == 0) {
      const float invS = 1.0f / (float)S_TOK;
      float acc = 0.0f;
      for (int e = 0; e < NEXP; ++e)
        acc += (sg[e] * invS) * ((float)sc[e] * invS);
      out[0] = acc * (float)NEXP;        // l_aux
    }
  }
}

// ---------------------------------------------------------------------------
extern "C" void kernel_launch(void* const* d_in, const int* in_sizes, int n_in,
                              void* d_out, int out_size, void* d_ws, size_t ws_size,
                              hipStream_t stream) {
  const float* inp = (const float*)d_in[0];   // [S, M] f32
  const float* wg  = (const float*)d_in[1];   // [E, M] f32
  float* out = (float*)d_out;

  // Workspace layout (~2.2 MB)
  char*  ws      = (char*)d_ws;
  float* logits  = (float*)ws;                                  // S*E f32
  int*   idx     = (int*)(ws + (size_t)S_TOK * NEXP * sizeof(float));
  int*   loc     = idx + S_TOK;
  float* gateval = (float*)(loc + S_TOK);
  float* pbsum   = gateval + S_TOK;                             // [32, 64] f32
  int*   pbcnt   = (int*)(pbsum + GATE_BLOCKS * NEXP);          // [32, 64] i32

  // 1) GEMM (WMMA f32): 8 waves/block, 1 wave = 16 tokens x all 64 experts
  k_gemm<<<S_TOK / (16 * 8), 256, 0, stream>>>(inp, wg, logits);

  // 2) softmax / argmax / partial reductions
  k_gate<<<GATE_BLOCKS, 256, 0, stream>>>(logits, idx, gateval, pbsum, pbcnt);

  // 3) per-expert cumsum positions (single workgroup, serial dependency)
  k_loc<<<1, 256, 0, stream>>>(idx, loc);

  // 4) zero-fill output (bandwidth-dominant: 536 MB)
  const long long SEC    = (long long)S_TOK * NEXP * CAP;
  const long long ntotal = 1 + 2 * SEC + NEXP;
  const long long n4     = ntotal / 4;
  const int fill_blocks  = (int)((n4 + 255) / 256);
  k_fill<<<fill_blocks, 256, 0, stream>>>((float4*)out, out, n4, ntotal);

  // 5) scatter nonzeros + exp_counts + l_aux
  k_finalize<<<GATE_BLOCKS, 256, 0, stream>>>(idx, loc, gateval, pbsum, pbcnt, out);
}